// TimeDomainBeamformer_46583215292559
// MI455X (gfx1250) — compile-verified
//
#include <hip/hip_runtime.h>
#include <hip/hip_bf16.h>

#define N_MICS   192
#define OVERLAP  1024
#define WINDOW   524288
#define ROWLEN   (OVERLAP + WINDOW)   // 525312 floats per mic row
#define NROWS    (2 * N_MICS)         // 384 (mic,tap) rows

typedef float v2f __attribute__((ext_vector_type(2)));
typedef float v8f __attribute__((ext_vector_type(8)));

// ---------------------------------------------------------------------------
// Kernel 1: per-mic geometry -> interpolation weights + row element offsets.
// Mirrors the reference math exactly (f32 sqrt, truncating int cast).
// wts[2m]  = 1-frac, wts[2m+1]  = frac
// offs[2m] = m*ROWLEN + OVERLAP - delay_int        (tap 0)
// offs[2m+1] = offs[2m] - 1                        (tap 1)
// ---------------------------------------------------------------------------
__global__ __launch_bounds__(N_MICS)
void beam_prep(const float* __restrict__ pos,      // 3 floats
               const float* __restrict__ mic_pos,  // 192*3 floats
               float* __restrict__ wts,            // 384 floats
               int* __restrict__ offs)             // 384 ints
{
    __shared__ float dist_s[N_MICS];
    __shared__ float dmax_s;

    const int m = threadIdx.x;
    const float dx = pos[0] - mic_pos[3 * m + 0];
    const float dy = pos[1] - mic_pos[3 * m + 1];
    const float dz = pos[2] - mic_pos[3 * m + 2];
    const float d  = sqrtf(dx * dx + dy * dy + dz * dz);
    dist_s[m] = d;
    __syncthreads();

    if (m == 0) {
        float mx = dist_s[0];
        for (int i = 1; i < N_MICS; ++i) mx = fmaxf(mx, dist_s[i]);
        dmax_s = mx;
    }
    __syncthreads();

    const float dd    = d - dmax_s;                              // <= 0
    const float delay = (-dd / 343000.0f) * 768000.0f / 16.0f;   // >= 0
    const int   di    = (int)delay;                              // trunc toward 0
    const float frac  = delay - (float)di;

    const int base = m * ROWLEN + OVERLAP - di;
    wts[2 * m + 0]  = 1.0f - frac;
    wts[2 * m + 1]  = frac;
    offs[2 * m + 0] = base;
    offs[2 * m + 1] = base - 1;
}

// ---------------------------------------------------------------------------
// Kernel 2: out(1xT) = w(1x384) * S(384xT) via V_WMMA_F32_16X16X4_F32.
// Each wave produces 16 consecutive output samples; K=384 accumulated in 96
// chained WMMAs. A = weights broadcast to all 16 rows (redundant FLOPs are
// free: kernel is HBM-bound at ~403MB / 23.3TB/s). B half-wave loads are
// 64B-contiguous. Uniform control flow -> EXEC all ones at every WMMA.
// ---------------------------------------------------------------------------
__global__ __launch_bounds__(256)
void beam_wmma(const float* __restrict__ buf,      // 192 x 525312
               const float* __restrict__ wts_g,    // 384
               const int* __restrict__ offs_g,     // 384
               float* __restrict__ out)            // 524288
{
    __shared__ float wts_s[NROWS];
    __shared__ int   offs_s[NROWS];
    for (int i = threadIdx.x; i < NROWS; i += 256) {
        wts_s[i]  = wts_g[i];
        offs_s[i] = offs_g[i];
    }
    __syncthreads();

    const int lane  = threadIdx.x & 31;
    const int wave  = threadIdx.x >> 5;
    const int t0    = (blockIdx.x * 8 + wave) * 16;   // 16 samples per wave
    const int col   = lane & 15;                      // N index 0..15
    const int half2 = (lane >> 4) << 1;               // K sub-select: 0 or 2
    const int t     = t0 + col;

    v8f acc = {};   // C/D accumulator, 16x16 f32 (rows all identical)

    #pragma unroll 4
    for (int k = 0; k < NROWS; k += 4) {
        v2f a, b;
        // A (16x4): every row holds the same 4 weights of this K-chunk.
        a.x = wts_s[k + half2 + 0];
        a.y = wts_s[k + half2 + 1];
        // B (4x16): 4 shifted buffer rows at 16 consecutive time samples.
        b.x = buf[offs_s[k + half2 + 0] + t];
        b.y = buf[offs_s[k + half2 + 1] + t];
        acc = __builtin_amdgcn_wmma_f32_16x16x4_f32(
                  /*neg_a=*/false, a, /*neg_b=*/false, b,
                  /*c_mod=*/(short)0, acc,
                  /*reuse_a=*/false, /*reuse_b=*/false);
    }

    // Row M=0 of D lives in acc[0] on lanes 0..15. Divide by 192 == mean.
    if (lane < 16) {
        out[t0 + lane] = acc[0] / 192.0f;
    }
}

// ---------------------------------------------------------------------------
// Launch wrapper.
// Inputs (reference order): pos (3), mic_pos (576), buffer (192*525312).
// d_ws layout: [0..383] float weights, [384..767] int32 offsets.
// ---------------------------------------------------------------------------
extern "C" void kernel_launch(void* const* d_in, const int* in_sizes, int n_in,
                              void* d_out, int out_size, void* d_ws, size_t ws_size,
                              hipStream_t stream) {
    const float* pos     = (const float*)d_in[0];
    const float* mic_pos = (const float*)d_in[1];
    const float* buf     = (const float*)d_in[2];
    float*       out     = (float*)d_out;

    float* wts  = (float*)d_ws;
    int*   offs = (int*)d_ws + NROWS;

    beam_prep<<<1, N_MICS, 0, stream>>>(pos, mic_pos, wts, offs);

    // WINDOW / (8 waves * 16 samples) = 524288 / 128 = 4096 blocks
    beam_wmma<<<WINDOW / 128, 256, 0, stream>>>(buf, wts, offs, out);
}